// FTP_Layer_17514876633823
// MI455X (gfx1250) — compile-verified
//
#include <hip/hip_runtime.h>

#define T_DIM 2048
#define K_DIM 64
#define MAX_NSEG 32
#define MIN_NSEG 8
#define DEV_INF __builtin_inff()

typedef float v2f __attribute__((ext_vector_type(2)));
typedef float v8f __attribute__((ext_vector_type(8)));

// ---------------------------------------------------------------------------
// 1) Xc[t,k] = cumsum(-X, axis=0). 64 threads, one per column; coalesced rows.
// ---------------------------------------------------------------------------
__global__ void k_cumsum(const float* __restrict__ X, float* __restrict__ Xc) {
    int k = threadIdx.x;
    float acc = 0.0f;
#pragma unroll 8
    for (int t = 0; t < T_DIM; ++t) {
        acc -= X[t * K_DIM + k];
        Xc[t * K_DIM + k] = acc;
    }
}

// ---------------------------------------------------------------------------
// 2) L[0][j] = (j>0) ? min_k Xc[j,k] : 0   (V[0,j] with zero shift row)
// ---------------------------------------------------------------------------
__global__ void k_l0(const float* __restrict__ Xc, float* __restrict__ L) {
    int j = blockIdx.x * 128 + threadIdx.x;
    const float4* row = (const float4*)(Xc + j * K_DIM);
    float m = DEV_INF;
#pragma unroll
    for (int q = 0; q < K_DIM / 4; ++q) {
        float4 v = row[q];
        m = fminf(m, fminf(fminf(v.x, v.y), fminf(v.z, v.w)));
    }
    L[j] = (j > 0) ? m : 0.0f;
}

// ---------------------------------------------------------------------------
// 3) Vs[m,j]: m==T-1 -> +inf ; j<=m+1 -> 0 ; else min_k(Xc[j,k]-Xc[m,k]).
//    64x64 tiles, Xc panels in LDS (pad 65 -> conflict-free), 16 outs/thread.
// ---------------------------------------------------------------------------
__global__ void k_vs(const float* __restrict__ Xc, float* __restrict__ Vs) {
    __shared__ float lm[64][65];
    __shared__ float lj[64][65];
    const int tid = threadIdx.x;         // 256
    const int m0 = blockIdx.y * 64;
    const int j0 = blockIdx.x * 64;
    const int tx = tid & 15;
    const int ty = tid >> 4;

    float acc[4][4];
#pragma unroll
    for (int a = 0; a < 4; ++a)
#pragma unroll
        for (int b = 0; b < 4; ++b) acc[a][b] = DEV_INF;

    const bool skip = (j0 + 63) <= (m0 + 1);   // tile fully in triu-zero region
    if (!skip) {
#pragma unroll
        for (int i = 0; i < 4; ++i) {          // 64 rows x 16 float4 per panel
            int f = tid + i * 256;
            int row = f >> 4;
            int q = (f & 15) * 4;
            float4 vm = *(const float4*)(Xc + (m0 + row) * K_DIM + q);
            float4 vj = *(const float4*)(Xc + (j0 + row) * K_DIM + q);
            lm[row][q + 0] = vm.x; lm[row][q + 1] = vm.y;
            lm[row][q + 2] = vm.z; lm[row][q + 3] = vm.w;
            lj[row][q + 0] = vj.x; lj[row][q + 1] = vj.y;
            lj[row][q + 2] = vj.z; lj[row][q + 3] = vj.w;
        }
        __syncthreads();
        for (int k = 0; k < K_DIM; ++k) {
            float am[4], bj[4];
#pragma unroll
            for (int a = 0; a < 4; ++a) am[a] = lm[ty * 4 + a][k];
#pragma unroll
            for (int b = 0; b < 4; ++b) bj[b] = lj[b * 16 + tx][k];
#pragma unroll
            for (int a = 0; a < 4; ++a)
#pragma unroll
                for (int b = 0; b < 4; ++b)
                    acc[a][b] = fminf(acc[a][b], bj[b] - am[a]);
        }
    }
#pragma unroll
    for (int a = 0; a < 4; ++a) {
        int m = m0 + ty * 4 + a;
#pragma unroll
        for (int b = 0; b < 4; ++b) {
            int j = j0 + b * 16 + tx;      // coalesced over tx
            float v = (m == T_DIM - 1) ? DEV_INF
                                       : ((j <= m + 1) ? 0.0f : acc[a][b]);
            Vs[m * T_DIM + j] = v;
        }
    }
}

// ---------------------------------------------------------------------------
// 4) DP step i: L[i][j] = (j>i) ? min_{m in [i, j-1]} (L[i-1][m] + Vs[m,j]) : 0
//    Stage A: partial mins over 8 m-chunks of 256 (grid parallelism).
// ---------------------------------------------------------------------------
__global__ void k_dp_partial(const float* __restrict__ Vs,
                             const float* __restrict__ Lprev,
                             float* __restrict__ part, int i) {
    __shared__ float prev[256];
    const int tid = threadIdx.x;            // 128
    const int jb = blockIdx.x;              // 0..15
    const int mc = blockIdx.y;              // 0..7
    const int mBase = mc * 256;
    for (int s = tid; s < 256; s += 128) prev[s] = Lprev[mBase + s];
    __syncthreads();
    const int j = jb * 128 + tid;
    const int jMax = jb * 128 + 127;
    int mLo = mBase > i ? mBase : i;
    int mHi = (mBase + 255) < (jMax - 1) ? (mBase + 255) : (jMax - 1);
    float acc = DEV_INF;
    for (int m = mLo; m <= mHi; ++m) {
        if (m <= j - 1)                      // coalesced Vs row reads over j
            acc = fminf(acc, prev[m - mBase] + Vs[m * T_DIM + j]);
    }
    part[mc * T_DIM + j] = acc;
}

__global__ void k_dp_combine(const float* __restrict__ part,
                             float* __restrict__ Lcur, int i) {
    const int j = blockIdx.x * 128 + threadIdx.x;
    float acc = DEV_INF;
#pragma unroll
    for (int mc = 0; mc < 8; ++mc) acc = fminf(acc, part[mc * T_DIM + j]);
    Lcur[j] = (j > i) ? acc : 0.0f;
}

// ---------------------------------------------------------------------------
// 5) opt_n_seg = argmin_n ( L[n-1][T-1] + n * T ),  n in [8, 32]
// ---------------------------------------------------------------------------
__global__ void k_select(const float* __restrict__ L, int* __restrict__ optn) {
    float best = DEV_INF;
    int bi = 0;
    for (int k = 0; k <= MAX_NSEG - MIN_NSEG; ++k) {
        float loss = L[(MIN_NSEG - 1 + k) * T_DIM + (T_DIM - 1)]
                   + (float)(MIN_NSEG + k) * (float)T_DIM;   // PENALTY_WEIGHT*T
        if (loss < best) { best = loss; bi = k; }            // first-min
    }
    optn[0] = bi + (MIN_NSEG - 1);
}

// ---------------------------------------------------------------------------
// 6) Backtrack: 32 sequential argmins over m<opt_t of L[ic-1][m]+Vs[m,opt_t].
//    First-occurrence tiebreak (matches jnp.argmin).
// ---------------------------------------------------------------------------
__global__ void k_backtrack(const float* __restrict__ L,
                            const float* __restrict__ Vs,
                            const int* __restrict__ optn,
                            int* __restrict__ is_cp) {
    __shared__ float sval[256];
    __shared__ int   sidx[256];
    const int tid = threadIdx.x;
    for (int t = tid; t < T_DIM; t += 256) is_cp[t] = (t == T_DIM - 1) ? 1 : 0;
    __syncthreads();
    const int opt_n = optn[0];
    int  opt_t = T_DIM - 1;
    bool done = false;
    for (int c = 0; c < MAX_NSEG; ++c) {
        int  i = opt_n - c;
        bool active = (i >= 1) && !done;
        int  ic = (i > 1) ? i : 1;
        const float* Lr = L + (ic - 1) * T_DIM;
        float bv = DEV_INF;
        int   bi = 0x7fffffff;
        for (int m = tid; m < T_DIM; m += 256) {
            float v = (m < opt_t) ? (Lr[m] + Vs[m * T_DIM + opt_t]) : DEV_INF;
            if (v < bv || (v == bv && m < bi)) { bv = v; bi = m; }
        }
        sval[tid] = bv; sidx[tid] = bi;
        __syncthreads();
        for (int s = 128; s > 0; s >>= 1) {
            if (tid < s) {
                float ov = sval[tid + s]; int oi = sidx[tid + s];
                if (ov < sval[tid] || (ov == sval[tid] && oi < sidx[tid])) {
                    sval[tid] = ov; sidx[tid] = oi;
                }
            }
            __syncthreads();
        }
        int cand = sidx[0];
        __syncthreads();
        if (active) {
            opt_t = cand;
            if (tid == 0) is_cp[cand] = 1;
        }
        done = done || (active && (cand == 0));
        __syncthreads();
    }
}

// ---------------------------------------------------------------------------
// 7) seg_id = exclusive prefix-sum of is_cp; counts = histogram(seg_id, 32)
// ---------------------------------------------------------------------------
__global__ void k_scan(const int* __restrict__ is_cp, int* __restrict__ seg,
                       float* __restrict__ counts) {
    __shared__ int partials[256];
    __shared__ int cnt[MAX_NSEG];
    const int tid = threadIdx.x;            // 256
    if (tid < MAX_NSEG) cnt[tid] = 0;
    int loc[8], s = 0;
    const int base = tid * 8;
#pragma unroll
    for (int q = 0; q < 8; ++q) { loc[q] = is_cp[base + q]; s += loc[q]; }
    partials[tid] = s;
    __syncthreads();
    if (tid == 0) {
        int run = 0;
        for (int k = 0; k < 256; ++k) { int t = partials[k]; partials[k] = run; run += t; }
    }
    __syncthreads();
    int run = partials[tid];
#pragma unroll
    for (int q = 0; q < 8; ++q) {
        seg[base + q] = run;
        if (run < MAX_NSEG) atomicAdd(&cnt[run], 1);
        run += loc[q];
    }
    __syncthreads();
    if (tid < MAX_NSEG) counts[tid] = (float)cnt[tid];
}

// ---------------------------------------------------------------------------
// 8) prototypes = (S @ X) / max(counts,1), S = one-hot(seg_id) (32 x 2048).
//    True f32 GEMM -> V_WMMA_F32_16X16X4_F32. 8 waves, one 16x16 tile each,
//    512 K-chunks of 4. Fragment layouts per ISA 7.12.2:
//      A (16x4): lane<16 -> M=lane, holds K={0,1}; lane>=16 -> M=lane-16, K={2,3}
//      B (4x16): lane%16 = N; v0 holds K={0|2}, v1 holds K={1|3} per lane-half
//      C/D: VGPR v -> M = v + 8*(lane>=16), N = lane%16
// ---------------------------------------------------------------------------
__global__ void k_proto(const float* __restrict__ X, const int* __restrict__ seg,
                        const float* __restrict__ counts, float* __restrict__ out) {
    __shared__ int sseg[T_DIM];
    const int tid = threadIdx.x;            // 256 = 8 waves
    for (int t = tid; t < T_DIM; t += 256) sseg[t] = seg[t];
    __syncthreads();
    const int wave = tid >> 5;
    const int lane = tid & 31;
    const int half = lane >> 4;
    const int lp   = lane & 15;
    const int mt = wave >> 2;               // 0..1  (M tile)
    const int nt = wave & 3;                // 0..3  (N tile)
    const int mRow = mt * 16 + lp;          // A indicator row
    const int nCol = nt * 16 + lp;          // B column

    v8f c = {};
    for (int kb = 0; kb < T_DIM / 4; ++kb) {
        const int t = kb * 4 + half * 2;
        v2f a, b;
        a.x = (sseg[t]     == mRow) ? 1.0f : 0.0f;
        a.y = (sseg[t + 1] == mRow) ? 1.0f : 0.0f;
        b.x = X[t * K_DIM + nCol];
        b.y = X[(t + 1) * K_DIM + nCol];
        c = __builtin_amdgcn_wmma_f32_16x16x4_f32(
                /*neg_a=*/false, a, /*neg_b=*/false, b,
                /*c_mod=*/(short)0, c, /*reuse_a=*/false, /*reuse_b=*/false);
    }
#pragma unroll
    for (int v = 0; v < 8; ++v) {
        int row = mt * 16 + v + half * 8;
        out[row * K_DIM + nCol] = c[v] / fmaxf(counts[row], 1.0f);
    }
}

// ---------------------------------------------------------------------------
extern "C" void kernel_launch(void* const* d_in, const int* in_sizes, int n_in,
                              void* d_out, int out_size, void* d_ws, size_t ws_size,
                              hipStream_t stream) {
    (void)in_sizes; (void)n_in; (void)out_size; (void)ws_size;
    const float* X = (const float*)d_in[0];
    float* out = (float*)d_out;
    char* ws = (char*)d_ws;

    const size_t OFF_XC   = 0;                                   // 512 KB
    const size_t OFF_VS   = OFF_XC + (size_t)T_DIM * K_DIM * 4;  // 16 MB
    const size_t OFF_L    = OFF_VS + (size_t)T_DIM * T_DIM * 4;  // 256 KB
    const size_t OFF_PART = OFF_L  + (size_t)MAX_NSEG * T_DIM * 4; // 64 KB
    const size_t OFF_ISCP = OFF_PART + (size_t)8 * T_DIM * 4;
    const size_t OFF_SEG  = OFF_ISCP + (size_t)T_DIM * 4;
    const size_t OFF_CNT  = OFF_SEG + (size_t)T_DIM * 4;
    const size_t OFF_OPTN = OFF_CNT + 128;

    float* Xc     = (float*)(ws + OFF_XC);
    float* Vs     = (float*)(ws + OFF_VS);
    float* L      = (float*)(ws + OFF_L);
    float* part   = (float*)(ws + OFF_PART);
    int*   iscp   = (int*)  (ws + OFF_ISCP);
    int*   seg    = (int*)  (ws + OFF_SEG);
    float* counts = (float*)(ws + OFF_CNT);
    int*   optn   = (int*)  (ws + OFF_OPTN);

    k_cumsum<<<1, 64, 0, stream>>>(X, Xc);
    k_l0<<<T_DIM / 128, 128, 0, stream>>>(Xc, L);
    k_vs<<<dim3(T_DIM / 64, T_DIM / 64), 256, 0, stream>>>(Xc, Vs);
    for (int i = 1; i < MAX_NSEG; ++i) {
        k_dp_partial<<<dim3(16, 8), 128, 0, stream>>>(Vs, L + (size_t)(i - 1) * T_DIM, part, i);
        k_dp_combine<<<16, 128, 0, stream>>>(part, L + (size_t)i * T_DIM, i);
    }
    k_select<<<1, 1, 0, stream>>>(L, optn);
    k_backtrack<<<1, 256, 0, stream>>>(L, Vs, optn, iscp);
    k_scan<<<1, 256, 0, stream>>>(iscp, seg, counts);
    k_proto<<<1, 256, 0, stream>>>(X, seg, counts, out);
}